// sparse3DBA_70076686402277
// MI455X (gfx1250) — compile-verified
//
#include <hip/hip_runtime.h>
#include <math.h>

#define N_PTS 20000
#define C_CH  256
#define H_IMG 480
#define W_PIX 640
#define HW    (H_IMG * W_PIX)
#define NB    ((N_PTS + 255) / 256)

typedef float v2f __attribute__((ext_vector_type(2)));
typedef float v8f __attribute__((ext_vector_type(8)));

// workspace float-slot layout
#define WS_R        0   // 9  current rotation (row-major)
#define WS_T        9   // 3  current translation
#define WS_LAM     12   // 1  lambda
#define WS_PREV    13   // 1  prev_cost
#define WS_RC      16   // 9  candidate rotation
#define WS_TC      25   // 3  candidate translation
#define WS_NAN     28   // 1  nan flag
#define WS_H       32   // 49 7x7 Gram: [0:6,0:6]=Hess, [:,6]=Grad
#define WS_COST    81   // 1  sum of err^2 (old pose)
#define WS_NEWCOST 82   // 1  sum of err^2 (candidate pose)
#define WS_OFF     96   // N ints: clamped pixel offset
#define WS_A (96 + N_PTS) // N*12 floats: A = J_e_p @ J_p_T (2x6 per point)

// ---------------------------------------------------------------- init / zero
__global__ void k_init(float* s) {
  int t = threadIdx.x;
  if (t < 9)  s[WS_R + t] = (t % 4 == 0) ? 1.f : 0.f;
  if (t == 0) { s[WS_T+0] = 1.f; s[WS_T+1] = 1.f; s[WS_T+2] = 0.f;
                s[WS_LAM] = 0.01f; s[WS_PREV] = 0.f; }
}
__global__ void k_zero(float* s) {
  int t = threadIdx.x;
  if (t < 51) s[WS_H + t] = 0.f;   // covers H[49] + COST + NEWCOST
}

// ------------------------------------------------- stage 1: project + A (2x6)
__global__ void __launch_bounds__(256)
k_project(const float* __restrict__ pts, const float* __restrict__ K, float* s) {
  int gid = blockIdx.x * blockDim.x + threadIdx.x;
  if (gid >= N_PTS) return;
  float X = pts[gid*3+0], Y = pts[gid*3+1], Z = pts[gid*3+2];
  const float* R = s + WS_R; const float* t = s + WS_T;
  float px = R[0]*X + R[1]*Y + R[2]*Z + t[0];
  float py = R[3]*X + R[4]*Y + R[5]*Z + t[1];
  float pz = R[6]*X + R[7]*Y + R[8]*Z + t[2];
  float u = px*K[0] + py*K[3] + pz*K[6];
  float v = px*K[1] + py*K[4] + pz*K[7];
  float w = px*K[2] + py*K[5] + pz*K[8];
  int i = (int)(u / w) - 1;               // trunc-toward-zero then -1 (ref)
  int j = (int)(v / w) - 1;
  i = min(max(i, 0), H_IMG - 1);
  j = min(max(j, 0), W_PIX - 1);
  ((int*)s)[WS_OFF + gid] = i * W_PIX + j;
  float iz = 1.0f / pz, iz2 = iz * iz;
  float* Ap = s + WS_A + (size_t)gid * 12;
  // row 0: (1/z)[1, 0, -x/z, -xy/z, z + x^2/z, -y]
  Ap[0] = iz;  Ap[1] = 0.f; Ap[2] = -px*iz2;
  Ap[3] = -px*py*iz2; Ap[4] = 1.f + px*px*iz2; Ap[5] = -py*iz;
  // row 1: (1/z)[0, 1, -y/z, -z - y^2/z, xy/z, x]
  Ap[6] = 0.f; Ap[7] = iz;  Ap[8] = -py*iz2;
  Ap[9] = -1.f - py*py*iz2; Ap[10] = px*py*iz2; Ap[11] = px*iz;
}

// -------------------- stage 2: gather + channel reduce + WMMA Gram (H & grad)
__global__ void __launch_bounds__(256)
k_reduce(const float* __restrict__ fmq, const float* __restrict__ fgx,
         const float* __restrict__ fgy, const float* __restrict__ ref,
         float* s) {
  __shared__ float lds[8 * 64 * 8];   // per-wave 64 rows x 8 floats
  int tid  = threadIdx.x;
  int gid  = blockIdx.x * blockDim.x + tid;
  int n    = gid < N_PTS ? gid : N_PTS - 1;   // clamp tail: keep EXEC full
  float wv = gid < N_PTS ? 1.f : 0.f;
  int off  = ((const int*)s)[WS_OFF + n];
  const float* refp = ref + (size_t)n * C_CH;

  float sxx=0.f, sxy=0.f, syy=0.f, sxe=0.f, sye=0.f, see=0.f;
  for (int c = 0; c < C_CH; ++c) {
    size_t b = (size_t)c * HW + off;
    float q = fmq[b], gx = fgx[b], gy = fgy[b];
    float e = q - refp[c];
    sxx = fmaf(gx, gx, sxx); sxy = fmaf(gx, gy, sxy); syy = fmaf(gy, gy, syy);
    sxe = fmaf(gx, e, sxe);  sye = fmaf(gy, e, sye);  see = fmaf(e, e, see);
    size_t bn = (size_t)((c + 1 < C_CH) ? c + 1 : c) * HW + off;
    __builtin_prefetch(fmq + bn, 0, 1);   // global_prefetch_b8
    __builtin_prefetch(fgx + bn, 0, 1);
    __builtin_prefetch(fgy + bn, 0, 1);
  }

  // 2x2 Cholesky of channel sums -> two 7-wide rows [B | u] per point
  const float eps = 1e-20f;
  float l00 = sqrtf(sxx + eps);
  float i00 = 1.f / l00;
  float l10 = sxy * i00;
  float t11 = syy - l10 * l10; t11 = t11 > 0.f ? t11 : 0.f;
  float l11 = sqrtf(t11 + eps);
  float u0  = sxe * i00;
  float u1  = (sye - l10 * u0) / l11;

  const float* Ap = s + WS_A + (size_t)n * 12;
  float r0[8], r1[8];
  #pragma unroll
  for (int m = 0; m < 6; ++m) {
    float a0 = Ap[m], a1 = Ap[6 + m];
    r0[m] = wv * (l00 * a0 + l10 * a1);
    r1[m] = wv * (l11 * a1);
  }
  r0[6] = wv * u0; r1[6] = wv * u1; r0[7] = 0.f; r1[7] = 0.f;

  int wave = tid >> 5, lane = tid & 31;
  float* wb = lds + wave * 512;
  #pragma unroll
  for (int m = 0; m < 8; ++m) {
    wb[(lane * 2 + 0) * 8 + m] = r0[m];
    wb[(lane * 2 + 1) * 8 + m] = r1[m];
  }
  __syncthreads();

  // Gram accumulation: 16 x v_wmma_f32_16x16x4_f32 per wave (64 rows, K=4/step)
  // A(m,k)=row_k[m] and B(k,n)=row_k[n] occupy identical per-lane slots.
  v8f acc = {0.f,0.f,0.f,0.f,0.f,0.f,0.f,0.f};
  int Kb = (lane >> 4) * 2;
  int M  = lane & 15;
  int Ms = M < 8 ? M : 0;
  for (int kc = 0; kc < 16; ++kc) {
    float x0 = wb[(kc * 4 + Kb + 0) * 8 + Ms];
    float x1 = wb[(kc * 4 + Kb + 1) * 8 + Ms];
    x0 = (M < 8) ? x0 : 0.f;
    x1 = (M < 8) ? x1 : 0.f;
    v2f op = {x0, x1};
    acc = __builtin_amdgcn_wmma_f32_16x16x4_f32(
        false, op, false, op, (short)0, acc, false, false);
  }
  // C/D layout: VGPR v, lanes 0-15 -> row M=v, col N=lane. Need rows/cols 0..6.
  if (lane < 7) {
    #pragma unroll
    for (int mr = 0; mr < 7; ++mr)
      atomicAdd(&s[WS_H + mr * 7 + lane], acc[mr]);
  }

  __syncthreads();
  lds[tid] = see * wv;                 // cost sum (err^2)
  __syncthreads();
  for (int st = 128; st > 0; st >>= 1) {
    if (tid < st) lds[tid] += lds[tid + st];
    __syncthreads();
  }
  if (tid == 0) atomicAdd(&s[WS_COST], lds[0]);
}

// ------------------------------------------ stage 3: LM solve + SO(3) update
__device__ inline void m3mul(const float* A, const float* B, float* C) {
  for (int i = 0; i < 3; ++i)
    for (int j = 0; j < 3; ++j)
      C[i*3+j] = A[i*3+0]*B[0*3+j] + A[i*3+1]*B[1*3+j] + A[i*3+2]*B[2*3+j];
}

__global__ void k_solve(float* s, int iter) {
  if (threadIdx.x != 0 || blockIdx.x != 0) return;
  if (iter == 0) s[WS_PREV] = 0.5f * s[WS_COST] / (float)N_PTS;
  float lam = s[WS_LAM];
  float M[6][7];
  for (int m = 0; m < 6; ++m) {
    for (int l = 0; l < 6; ++l) M[m][l] = s[WS_H + m*7 + l];
    M[m][6] = s[WS_H + m*7 + 6];                 // gradient
    M[m][m] += (M[m][m] + 1e-9f) * lam;          // LM damping
  }
  // Gaussian elimination, partial pivot
  for (int col = 0; col < 6; ++col) {
    int piv = col; float best = fabsf(M[col][col]);
    for (int r = col + 1; r < 6; ++r) {
      float a = fabsf(M[r][col]); if (a > best) { best = a; piv = r; }
    }
    for (int l = col; l < 7; ++l) {
      float tmp = M[col][l]; M[col][l] = M[piv][l]; M[piv][l] = tmp;
    }
    float ip = 1.f / M[col][col];
    for (int r = col + 1; r < 6; ++r) {
      float f = M[r][col] * ip;
      for (int l = col; l < 7; ++l) M[r][l] -= f * M[col][l];
    }
  }
  float x[6], delta[6];
  for (int r = 5; r >= 0; --r) {
    float v = M[r][6];
    for (int l = r + 1; l < 6; ++l) v -= M[r][l] * x[l];
    x[r] = v / M[r][r];
  }
  int nan = 0;
  for (int m = 0; m < 6; ++m) { delta[m] = -x[m]; nan |= (delta[m] != delta[m]); }
  float dt[3] = {delta[0], delta[1], delta[2]};
  float dw[3] = {delta[3], delta[4], delta[5]};
  // so3exp
  float th2 = dw[0]*dw[0] + dw[1]*dw[1] + dw[2]*dw[2];
  float th  = sqrtf(th2 + 1e-24f);
  float Ac  = th < 1e-6f ? 1.f : sinf(th) / th;
  float Bc  = th < 1e-6f ? 0.5f : (1.f - cosf(th)) / th2;
  float Wm[9] = {0.f,-dw[2],dw[1], dw[2],0.f,-dw[0], -dw[1],dw[0],0.f};
  float W2[9]; m3mul(Wm, Wm, W2);
  float dr[9];
  for (int k = 0; k < 9; ++k)
    dr[k] = ((k % 4 == 0) ? 1.f : 0.f) + Ac * Wm[k] + Bc * W2[k];
  float Rc[9]; m3mul(dr, s + WS_R, Rc);
  const float* t0 = s + WS_T;
  for (int k = 0; k < 9; ++k) s[WS_RC + k] = Rc[k];
  for (int i = 0; i < 3; ++i)
    s[WS_TC + i] = dr[i*3+0]*t0[0] + dr[i*3+1]*t0[1] + dr[i*3+2]*t0[2] + dt[i];
  s[WS_NAN] = nan ? 1.f : 0.f;
  s[WS_NEWCOST] = 0.f;
}

// --------------------------------- stage 4: candidate-pose cost (no -1 offset)
__global__ void __launch_bounds__(256)
k_newcost(const float* __restrict__ pts, const float* __restrict__ K,
          const float* __restrict__ fmq, const float* __restrict__ ref, float* s) {
  __shared__ float red[256];
  int tid = threadIdx.x;
  int gid = blockIdx.x * blockDim.x + tid;
  int n   = gid < N_PTS ? gid : N_PTS - 1;
  float wv = gid < N_PTS ? 1.f : 0.f;
  const float* R = s + WS_RC; const float* t = s + WS_TC;
  float X = pts[n*3+0], Y = pts[n*3+1], Z = pts[n*3+2];
  float px = R[0]*X + R[1]*Y + R[2]*Z + t[0];
  float py = R[3]*X + R[4]*Y + R[5]*Z + t[1];
  float pz = R[6]*X + R[7]*Y + R[8]*Z + t[2];
  float u = px*K[0] + py*K[3] + pz*K[6];
  float v = px*K[1] + py*K[4] + pz*K[7];
  float w = px*K[2] + py*K[5] + pz*K[8];
  int i = (int)(u / w);                         // NOTE: no -1 here (matches ref)
  int j = (int)(v / w);
  i = min(max(i, 0), H_IMG - 1);
  j = min(max(j, 0), W_PIX - 1);
  int off = i * W_PIX + j;
  const float* refp = ref + (size_t)n * C_CH;
  float sum = 0.f;
  for (int c = 0; c < C_CH; ++c) {
    float e = fmq[(size_t)c * HW + off] - refp[c];
    sum = fmaf(e, e, sum);
    size_t bn = (size_t)((c + 1 < C_CH) ? c + 1 : c) * HW + off;
    __builtin_prefetch(fmq + bn, 0, 1);
  }
  red[tid] = sum * wv;
  __syncthreads();
  for (int st = 128; st > 0; st >>= 1) {
    if (tid < st) red[tid] += red[tid + st];
    __syncthreads();
  }
  if (tid == 0) atomicAdd(&s[WS_NEWCOST], red[0]);
}

// --------------------------------------------- stage 5: accept/reject, lambda
__global__ void k_accept(float* s) {
  if (threadIdx.x != 0 || blockIdx.x != 0) return;
  float nc   = s[WS_NEWCOST] / (float)N_PTS;
  float prev = s[WS_PREV];
  int increased = nc > prev;                    // NaN compares false, like jnp
  float lam = s[WS_LAM] * (increased ? 10.f : 0.1f);
  s[WS_LAM] = fminf(fmaxf(lam, 1e-6f), 100.f);
  int acc = (!increased) && (s[WS_NAN] == 0.f);
  if (acc) {
    for (int k = 0; k < 9; ++k) s[WS_R + k] = s[WS_RC + k];
    for (int k = 0; k < 3; ++k) s[WS_T + k] = s[WS_TC + k];
    s[WS_PREV] = nc;
  }
}

__global__ void k_out(const float* s, float* out) {
  int t = threadIdx.x;
  if (t < 9) out[t] = s[WS_R + t];
  else if (t < 12) out[t] = s[WS_T + t - 9];
}

// ---------------------------------------------------------------------- host
extern "C" void kernel_launch(void* const* d_in, const int* in_sizes, int n_in,
                              void* d_out, int out_size, void* d_ws, size_t ws_size,
                              hipStream_t stream) {
  const float* pts = (const float*)d_in[0];
  const float* ref = (const float*)d_in[1];
  const float* fmq = (const float*)d_in[2];
  const float* fgx = (const float*)d_in[3];
  const float* fgy = (const float*)d_in[4];
  const float* K   = (const float*)d_in[5];
  float* s   = (float*)d_ws;
  float* out = (float*)d_out;

  k_init<<<1, 64, 0, stream>>>(s);
  for (int it = 0; it < 5; ++it) {
    k_zero<<<1, 64, 0, stream>>>(s);
    k_project<<<NB, 256, 0, stream>>>(pts, K, s);
    k_reduce<<<NB, 256, 0, stream>>>(fmq, fgx, fgy, ref, s);
    k_solve<<<1, 1, 0, stream>>>(s, it);
    k_newcost<<<NB, 256, 0, stream>>>(pts, K, fmq, ref, s);
    k_accept<<<1, 1, 0, stream>>>(s);
  }
  k_out<<<1, 32, 0, stream>>>(s, out);
}